// DQFMLoss_68719477146
// MI455X (gfx1250) — compile-verified
//
#include <hip/hip_runtime.h>
#include <math.h>

// Problem constants (from reference setup_inputs)
#define B_      2
#define M_      5000
#define C_      128
#define KS_     50
#define N_      2000
#define KNN_    1000
#define NCHUNK_ 400          // N processed per chunk; d2 scratch = B*NCHUNK*M floats = 16 MB
#define EPS_    1e-8f

typedef __attribute__((ext_vector_type(2))) float v2f;
typedef __attribute__((ext_vector_type(8))) float v8f;

// ---------------------------------------------------------------------------
// bb[row] = sum_c feat[row][c]^2 for feat1 (rows 0..B*M-1) and feat2 (next B*M)
// one wave32 per row
// ---------------------------------------------------------------------------
__global__ void row_norms_kernel(const float* __restrict__ feat1,
                                 const float* __restrict__ feat2,
                                 float* __restrict__ bb) {
  int wid  = (blockIdx.x * blockDim.x + threadIdx.x) >> 5;
  int lane = threadIdx.x & 31;
  if (wid >= 2 * B_ * M_) return;
  const float* feat = (wid < B_ * M_) ? feat1 : feat2;
  int row = wid % (B_ * M_);
  const float* p = feat + (size_t)row * C_;
  float s = 0.f;
  for (int c = lane; c < C_; c += 32) { float v = p[c]; s += v * v; }
  for (int off = 16; off > 0; off >>= 1) s += __shfl_down(s, off, 32);
  if (lane == 0) bb[wid] = s;
}

// ---------------------------------------------------------------------------
// d2[b, n, m] = aa[b,n] + bb[b,m] - 2 * dot(feat[b, ridx[n]], feat[b, m])
// One wave computes a 16x64 tile (4 accumulators) via V_WMMA_F32_16X16X4_F32;
// each A-fragment load is reused by 4 WMMAs (4:5 wmma:load steady state).
// Block = 256 threads = 8 waves covering 512 consecutive m columns.
// grid = (ceil(M/512)=10, NCHUNK/16=25, B)
// ---------------------------------------------------------------------------
__global__ void gemm_d2_kernel(const float* __restrict__ feat,
                               const int*  __restrict__ ridx,
                               const float* __restrict__ bb,
                               float* __restrict__ d2w,
                               int chunk_base) {
  const int lane = threadIdx.x & 31;
  const int wave = threadIdx.x >> 5;
  const int b    = blockIdx.z;
  const int m0   = (blockIdx.x * 8 + wave) * 64;   // 4 tiles of 16 columns
  const int n0   = chunk_base + blockIdx.y * 16;   // global sample-tile base
  const int half = lane >> 4;                      // 0: K pair {0,1}, 1: K pair {2,3}
  const int l15  = lane & 15;

  // A fragment (16x4): row n0+l15 of f1 = feat[b, ridx[n0+l15], :]
  const int na = n0 + l15;
  const float* pa = feat + ((size_t)b * M_ + ridx[na]) * C_ + 2 * half;

  // B fragments (4x16): columns m0+16t+l15 of feat^T = feat[b, m, :]
  const float* pb[4];
#pragma unroll
  for (int t = 0; t < 4; ++t) {
    int mb = m0 + 16 * t + l15;
    if (mb > M_ - 1) mb = M_ - 1;                  // clamp; guarded at store
    pb[t] = feat + ((size_t)b * M_ + mb) * C_ + 2 * half;
  }

  v8f acc0 = {}, acc1 = {}, acc2 = {}, acc3 = {};
  for (int k = 0; k < C_; k += 4) {
    v2f av = *(const v2f*)(pa + k);
    v2f b0 = *(const v2f*)(pb[0] + k);
    v2f b1 = *(const v2f*)(pb[1] + k);
    v2f b2 = *(const v2f*)(pb[2] + k);
    v2f b3 = *(const v2f*)(pb[3] + k);
    acc0 = __builtin_amdgcn_wmma_f32_16x16x4_f32(false, av, false, b0,
                                                 (short)0, acc0, false, false);
    acc1 = __builtin_amdgcn_wmma_f32_16x16x4_f32(false, av, false, b1,
                                                 (short)0, acc1, false, false);
    acc2 = __builtin_amdgcn_wmma_f32_16x16x4_f32(false, av, false, b2,
                                                 (short)0, acc2, false, false);
    acc3 = __builtin_amdgcn_wmma_f32_16x16x4_f32(false, av, false, b3,
                                                 (short)0, acc3, false, false);
  }

  // Epilogue.  D layout: vgpr r -> row r (lanes 0-15) / row r+8 (lanes 16-31)
  float aav[8];
#pragma unroll
  for (int r = 0; r < 8; ++r)
    aav[r] = bb[(size_t)b * M_ + ridx[n0 + r + 8 * half]];

  v8f accs[4] = {acc0, acc1, acc2, acc3};
#pragma unroll
  for (int t = 0; t < 4; ++t) {
    const int m = m0 + 16 * t + l15;
    if (m < M_) {
      const float bbm = bb[(size_t)b * M_ + m];
#pragma unroll
      for (int r = 0; r < 8; ++r) {
        const int nl = (n0 - chunk_base) + r + 8 * half;
        d2w[((size_t)b * NCHUNK_ + nl) * M_ + m] = aav[r] + bbm - 2.0f * accs[t][r];
      }
    }
  }
}

// ---------------------------------------------------------------------------
// Per-row (one workgroup): radix-select the 1000 smallest d2 of 5000, gather
// dist[b, idx, ridx[n]], compute sim = 1 - |<dr,df>| / (||dr|| ||df||).
// Row cached in LDS (20 KB). Fully deterministic (no float atomics).
// ---------------------------------------------------------------------------
__device__ __forceinline__ float block_sum256(float v, float* s_red) {
  s_red[threadIdx.x] = v;
  __syncthreads();
  for (int s = 128; s > 0; s >>= 1) {
    if ((int)threadIdx.x < s) s_red[threadIdx.x] += s_red[threadIdx.x + s];
    __syncthreads();
  }
  float r = s_red[0];
  __syncthreads();
  return r;
}

__global__ void select_sim_kernel(const float* __restrict__ d2w,
                                  const float* __restrict__ dist,
                                  const int*  __restrict__ ridx,
                                  float* __restrict__ simw,
                                  int chunk_base) {
  __shared__ float    s_row[M_];
  __shared__ unsigned s_hist[256];
  __shared__ float    s_red[256];
  __shared__ unsigned s_prefix;
  __shared__ int      s_rem;

  const int tid = threadIdx.x;
  const int b   = blockIdx.x / NCHUNK_;
  const int nl  = blockIdx.x % NCHUNK_;
  const int n   = chunk_base + nl;
  const float* row = d2w + ((size_t)b * NCHUNK_ + nl) * M_;

  for (int i = tid; i < M_; i += 256) s_row[i] = row[i];
  __syncthreads();

  // MSB-first 8-bit radix select of the KNN-th smallest key.
  // key = bits of max(d2,0) -- monotone for non-negative floats.
  unsigned prefix = 0, maskhi = 0;
  int rem = KNN_;
  for (int shift = 24; shift >= 0; shift -= 8) {
    s_hist[tid] = 0;
    __syncthreads();
    for (int i = tid; i < M_; i += 256) {
      unsigned k = __float_as_uint(fmaxf(s_row[i], 0.0f));
      if ((k & maskhi) == prefix) atomicAdd(&s_hist[(k >> shift) & 255], 1u);
    }
    __syncthreads();
    if (tid == 0) {
      int cum = 0, d = 0;
      for (; d < 256; ++d) {
        int h = (int)s_hist[d];
        if (cum + h >= rem) break;
        cum += h;
      }
      s_prefix = prefix | ((unsigned)d << shift);
      s_rem    = rem - cum;
    }
    __syncthreads();
    prefix = s_prefix;
    rem    = s_rem;
    maskhi |= (0xFFu << shift);
    __syncthreads();
  }
  const unsigned T = prefix;       // exact key of the KNN-th smallest
  const int ridx_n = ridx[n];

  float srd = 0.f, srr = 0.f, sff = 0.f;
  for (int i = tid; i < M_; i += 256) {
    unsigned k = __float_as_uint(fmaxf(s_row[i], 0.0f));
    bool sel = (k < T);
    if (!sel && k == T) {
      // deterministic tie-break: rank among equals by index
      int rank = 0;
      for (int j = 0; j < i; ++j)
        if (__float_as_uint(fmaxf(s_row[j], 0.0f)) == T) ++rank;
      sel = (rank < rem);
    }
    if (sel) {
      float dr = sqrtf(fmaxf(s_row[i], 1e-12f));
      float df = dist[((size_t)b * M_ + i) * M_ + ridx_n];
      srd += dr * df;
      srr += dr * dr;
      sff += df * df;
    }
  }
  float srd_t = block_sum256(srd, s_red);
  float srr_t = block_sum256(srr, s_red);
  float sff_t = block_sum256(sff, s_red);
  if (tid == 0) {
    float den = fmaxf(sqrtf(srr_t), EPS_) * fmaxf(sqrtf(sff_t), EPS_);
    simw[b * N_ + n] = 1.0f - fabsf(srd_t / den);
  }
}

// ---------------------------------------------------------------------------
// Finalize: small 50x50 Frobenius terms + sum of per-row sims -> 5 outputs.
// Single workgroup, deterministic tree reductions.
// ---------------------------------------------------------------------------
__device__ float frob_xxt_partial(const float* __restrict__ X, int tid) {
  float acc = 0.f;
  for (int e = tid; e < B_ * KS_ * KS_; e += 256) {
    int b = e / (KS_ * KS_), r = e % (KS_ * KS_);
    int i = r / KS_, j = r % KS_;
    const float* Xb = X + (size_t)b * KS_ * KS_;
    float s = 0.f;
    for (int k = 0; k < KS_; ++k) s += Xb[i * KS_ + k] * Xb[j * KS_ + k];
    float t = s - (i == j ? 1.0f : 0.0f);
    acc += t * t;
  }
  return acc;
}

__device__ float frob_ab_partial(const float* __restrict__ A,
                                 const float* __restrict__ Bm, int tid) {
  float acc = 0.f;
  for (int e = tid; e < B_ * KS_ * KS_; e += 256) {
    int b = e / (KS_ * KS_), r = e % (KS_ * KS_);
    int i = r / KS_, j = r % KS_;
    const float* Ab = A  + (size_t)b * KS_ * KS_;
    const float* Bb = Bm + (size_t)b * KS_ * KS_;
    float s = 0.f;
    for (int k = 0; k < KS_; ++k) s += Ab[i * KS_ + k] * Bb[k * KS_ + j];
    float t = s - (i == j ? 1.0f : 0.0f);
    acc += t * t;
  }
  return acc;
}

__device__ float frob_diff_partial(const float* __restrict__ A,
                                   const float* __restrict__ Bm, int tid) {
  float acc = 0.f;
  for (int e = tid; e < B_ * KS_ * KS_; e += 256) {
    float t = A[e] - Bm[e];
    acc += t * t;
  }
  return acc;
}

__global__ void finalize_kernel(const float* __restrict__ C12,
                                const float* __restrict__ C21,
                                const float* __restrict__ C12n,
                                const float* __restrict__ C21n,
                                const float* __restrict__ simw,
                                float* __restrict__ out) {
  __shared__ float s_red[256];
  const int tid = threadIdx.x;

  float p = 0.f;
  for (int i = tid; i < 2 * B_ * N_; i += 256) p += simw[i];
  float dist_raw = block_sum256(p, s_red);

  float o = frob_xxt_partial(C12, tid) + frob_xxt_partial(C21, tid) +
            frob_xxt_partial(C12n, tid) + frob_xxt_partial(C21n, tid);
  float ortho_raw = block_sum256(o, s_red);

  float bj = frob_ab_partial(C12, C21, tid) + frob_ab_partial(C21, C12, tid);
  float bij_raw = block_sum256(bj, s_red);

  float rs = frob_diff_partial(C12, C12n, tid) + frob_diff_partial(C21, C21n, tid);
  float res_raw = block_sum256(rs, s_red);

  if (tid == 0) {
    float dist_loss  = dist_raw * 0.5f;                   // (br1+br2)*w_dist/2
    float ortho_loss = ortho_raw * (0.5f / (float)B_);    // mean over B, *1/2
    float bij_loss   = bij_raw * (1.0f / (float)B_);
    float res_loss   = res_raw * (1.0f / (float)B_);
    out[0] = dist_loss + ortho_loss + bij_loss + res_loss;
    out[1] = ortho_loss;
    out[2] = bij_loss;
    out[3] = res_loss;
    out[4] = dist_loss;
  }
}

// ---------------------------------------------------------------------------
extern "C" void kernel_launch(void* const* d_in, const int* in_sizes, int n_in,
                              void* d_out, int out_size, void* d_ws, size_t ws_size,
                              hipStream_t stream) {
  (void)in_sizes; (void)n_in; (void)out_size; (void)ws_size;
  const float* C12   = (const float*)d_in[0];
  const float* C21   = (const float*)d_in[1];
  const float* C12n  = (const float*)d_in[2];
  const float* C21n  = (const float*)d_in[3];
  const float* feat1 = (const float*)d_in[4];
  const float* feat2 = (const float*)d_in[5];
  // d_in[6], d_in[7] (evecs_trans*) are unused by the reference
  const float* dist1 = (const float*)d_in[8];
  const float* dist2 = (const float*)d_in[9];
  const int* ridx1   = (const int*)d_in[10];
  const int* ridx2   = (const int*)d_in[11];
  float* out = (float*)d_out;

  // workspace layout (floats):
  //   [0, 20000)                      bb (feat1 rows then feat2 rows)
  //   [20000, 28000)                  sim values (branch-major, B*N each)
  //   [28000, 28000 + B*NCHUNK*M)     d2 chunk scratch (16 MB)
  float* ws   = (float*)d_ws;
  float* bb   = ws;
  float* simw = ws + 20000;
  float* d2w  = ws + 28000;

  row_norms_kernel<<<2500, 256, 0, stream>>>(feat1, feat2, bb);

  for (int br = 0; br < 2; ++br) {
    const float* feat = br ? feat2 : feat1;
    const float* dist = br ? dist2 : dist1;
    const int*   ridx = br ? ridx2 : ridx1;
    const float* bbr  = bb + (size_t)br * B_ * M_;
    float* simb = simw + (size_t)br * B_ * N_;
    for (int cb = 0; cb < N_; cb += NCHUNK_) {
      dim3 gg((M_ + 511) / 512, NCHUNK_ / 16, B_);
      gemm_d2_kernel<<<gg, 256, 0, stream>>>(feat, ridx, bbr, d2w, cb);
      select_sim_kernel<<<B_ * NCHUNK_, 256, 0, stream>>>(d2w, dist, ridx, simb, cb);
    }
  }

  finalize_kernel<<<1, 256, 0, stream>>>(C12, C21, C12n, C21n, simw, out);
}